// LDAMLoss_14319420965530
// MI455X (gfx1250) — compile-verified
//
#include <hip/hip_runtime.h>

// ---- problem constants (from reference) ----
#define S_SCALE   30.0f
#define NCLASS    32000
#define NBATCH    4096
#define L2E       1.44269504088896340736f   // log2(e)

// ---- launch shape: 32000 = 320 threads * 4 floats * 25 iters (exact) ----
#define TPB             320
#define NWAVES          (TPB / 32)
#define VEC_PER_THREAD  25

typedef __attribute__((ext_vector_type(2))) float v2f;
typedef __attribute__((ext_vector_type(8))) float v8f;

// Per-row LDAM NLL. One workgroup per batch row.
__global__ __launch_bounds__(TPB) void ldam_row_kernel(
    const float* __restrict__ logits,
    const int*   __restrict__ targets,
    const float* __restrict__ margins,
    const float* __restrict__ cweights,
    float*       __restrict__ per_row)   // d_ws: NBATCH floats
{
    const int row  = blockIdx.x;
    const int tid  = threadIdx.x;
    const int lane = tid & 31;
    const int wave = tid >> 5;

    const float* rp   = logits + (size_t)row * NCLASS;
    const float4* rp4 = (const float4*)rp;      // row base is 16B aligned (32000*4 % 16 == 0)

    __shared__ float s_red[NWAVES];
    __shared__ float s_bcast;

    // ---------- pass 1: max over raw logits (max(u) = S * max(x), S > 0) ----------
    float m = -3.402823466e+38f;
    #pragma unroll 5
    for (int i = 0; i < VEC_PER_THREAD; ++i) {
        float4 x = rp4[tid + i * TPB];
        m = fmaxf(m, fmaxf(fmaxf(x.x, x.y), fmaxf(x.z, x.w)));
    }
    #pragma unroll
    for (int off = 16; off > 0; off >>= 1)
        m = fmaxf(m, __shfl_xor(m, off, 32));
    if (lane == 0) s_red[wave] = m;
    __syncthreads();
    if (tid == 0) {
        float bm = s_red[0];
        #pragma unroll
        for (int w = 1; w < NWAVES; ++w) bm = fmaxf(bm, s_red[w]);
        s_bcast = bm;
    }
    __syncthreads();

    const float M = s_bcast * S_SCALE;      // shift in the scaled domain
    // exp(S*x - M) == exp2(x * (S*log2e) + (-M*log2e))  -> 1 fma + 1 v_exp_f32
    const float ca = S_SCALE * L2E;
    const float cb = -M * L2E;

    // ---------- pass 2: sum of exp via V_WMMA_F32_16X16X4_F32 (B = ones) ----------
    // Two independent accumulators break the WMMA->WMMA RAW chain so consecutive
    // v_wmma ops can pipeline on the matrix pipe (no accumulator serialization).
    v8f acc0 = {0.f, 0.f, 0.f, 0.f, 0.f, 0.f, 0.f, 0.f};
    v8f acc1 = {0.f, 0.f, 0.f, 0.f, 0.f, 0.f, 0.f, 0.f};
    v2f ones; ones.x = 1.0f; ones.y = 1.0f;
    #pragma unroll 5
    for (int i = 0; i < VEC_PER_THREAD; ++i) {
        float4 x = rp4[tid + i * TPB];
        v2f e01, e23;
        e01.x = __builtin_amdgcn_exp2f(__builtin_fmaf(x.x, ca, cb));
        e01.y = __builtin_amdgcn_exp2f(__builtin_fmaf(x.y, ca, cb));
        e23.x = __builtin_amdgcn_exp2f(__builtin_fmaf(x.z, ca, cb));
        e23.y = __builtin_amdgcn_exp2f(__builtin_fmaf(x.w, ca, cb));
        acc0 = __builtin_amdgcn_wmma_f32_16x16x4_f32(
                   false, e01, false, ones, (short)0, acc0, false, false);
        acc1 = __builtin_amdgcn_wmma_f32_16x16x4_f32(
                   false, e23, false, ones, (short)0, acc1, false, false);
    }
    // D[m][n] = rowsum_m (replicated across n). Per-lane sum of the 8 acc VGPRs
    // gives (lanes 0-15): sum of rowsums m=0..7, (lanes 16-31): m=8..15.
    float psum = (acc0[0] + acc1[0]) + (acc0[1] + acc1[1])
               + (acc0[2] + acc1[2]) + (acc0[3] + acc1[3])
               + (acc0[4] + acc1[4]) + (acc0[5] + acc1[5])
               + (acc0[6] + acc1[6]) + (acc0[7] + acc1[7]);
    float wsum = __shfl(psum, 0, 32) + __shfl(psum, 16, 32);
    if (lane == 0) s_red[wave] = wsum;
    __syncthreads();

    if (tid == 0) {
        float ssum = 0.f;
        #pragma unroll
        for (int w = 0; w < NWAVES; ++w) ssum += s_red[w];
        // margin correction: only the target element differs from u
        const int   t  = targets[row];
        const float ut = rp[t] * S_SCALE;
        const float zt = ut - S_SCALE * margins[t];
        ssum += __builtin_amdgcn_exp2f((zt - M) * L2E)
              - __builtin_amdgcn_exp2f((ut - M) * L2E);
        // nll = log(sum exp(z - M)) + M - zt   (v_log_f32 is log2)
        const float nll = __builtin_amdgcn_logf(ssum) * (1.0f / L2E) + M - zt;
        per_row[row] = cweights[row] * nll;
    }
}

// Deterministic mean over NBATCH per-row losses (single block).
__global__ __launch_bounds__(256) void reduce_mean_kernel(
    const float* __restrict__ per_row, float* __restrict__ out)
{
    __shared__ float s[8];
    const int tid = threadIdx.x;
    float v = 0.f;
    for (int i = tid; i < NBATCH; i += 256) v += per_row[i];
    #pragma unroll
    for (int off = 16; off > 0; off >>= 1) v += __shfl_xor(v, off, 32);
    if ((tid & 31) == 0) s[tid >> 5] = v;
    __syncthreads();
    if (tid == 0) {
        float t = 0.f;
        #pragma unroll
        for (int w = 0; w < 8; ++w) t += s[w];
        out[0] = t * (1.0f / NBATCH);
    }
}

extern "C" void kernel_launch(void* const* d_in, const int* in_sizes, int n_in,
                              void* d_out, int out_size, void* d_ws, size_t ws_size,
                              hipStream_t stream) {
    const float* logits   = (const float*)d_in[0];
    const int*   targets  = (const int*)d_in[1];
    const float* margins  = (const float*)d_in[2];
    const float* cweights = (const float*)d_in[3];
    float* out     = (float*)d_out;
    float* per_row = (float*)d_ws;                 // NBATCH floats of scratch

    ldam_row_kernel<<<NBATCH, TPB, 0, stream>>>(logits, targets, margins, cweights, per_row);
    reduce_mean_kernel<<<1, 256, 0, stream>>>(per_row, out);
}